// GPUAcceleratedGNN_20143396618495
// MI455X (gfx1250) — compile-verified
//
#include <hip/hip_runtime.h>

typedef __attribute__((ext_vector_type(2))) float v2f;
typedef __attribute__((ext_vector_type(8))) float v8f;

#define BN_EPS 1e-5f

// ---------------------------------------------------------------------------
// Utility kernels
// ---------------------------------------------------------------------------
__global__ void zero_f32v4(float4* __restrict__ p, int n4) {
    int i = blockIdx.x * blockDim.x + threadIdx.x;
    if (i < n4) p[i] = make_float4(0.f, 0.f, 0.f, 0.f);
}

__global__ void deg_count(const int* __restrict__ dst, float* __restrict__ deg, int E) {
    int e = blockIdx.x * blockDim.x + threadIdx.x;
    if (e < E) atomicAdd(&deg[dst[e]], 1.0f);
}

__global__ void make_dinv(float* __restrict__ d, int n) {
    int i = blockIdx.x * blockDim.x + threadIdx.x;
    if (i < n) d[i] = rsqrtf(d[i] + 1.0f);  // +1 self-loop
}

// ---------------------------------------------------------------------------
// FP32 WMMA GEMM: C[n,Ncol] = A[n,K] * B[K,Ncol]
//  - block = 256 threads = 8 waves; all waves share one 64-column strip (tn)
//  - B staged through LDS in K-chunks of 128 rows, stored PAIR-INTERLEAVED:
//    (B[2r][c], B[2r+1][c]) contiguous -> each lane's B fragment is a single
//    aligned ds_load_b64 (no register shuffles).
//    Pair-row stride 160 dwords == 32 (mod 64) -> half-wave reads use
//    disjoint bank ranges (lanes 0-15: banks 2m..2m+1, lanes 16-31: +32).
//  - each wave computes a 16x64 output strip: 4 accumulators,
//    V_WMMA_F32_16X16X4_F32, A fragment reused across the 4 N-tiles
// Requirements: n % 16 == 0, K % 128 == 0, Ncol % 64 == 0.
// ---------------------------------------------------------------------------
#define KCHUNK    128
#define PAIRROWS  (KCHUNK / 2)    // 64
#define PRSTRIDE  160             // dwords per pair-row (64 pairs + 16 pad pairs)

__global__ void gemm_wmma_f32(const float* __restrict__ A, const float* __restrict__ B,
                              float* __restrict__ C, int n, int K, int Ncol) {
    __shared__ float ldsB[PAIRROWS * PRSTRIDE];   // 40,960 bytes

    const int tid  = threadIdx.x;
    const int lane = tid & 31;
    const int wid  = tid >> 5;                 // wave id in block (0..7)
    const int colStrips = Ncol >> 6;           // 64-column strips
    const int rowTiles  = n >> 4;
    const int tn       = blockIdx.x % colStrips;
    const int blockRow = blockIdx.x / colStrips;
    const int tm       = blockRow * 8 + wid;
    const bool valid   = (tm < rowTiles);      // wave-uniform

    const int m    = lane & 15;                // A row in tile / B,C column in tile
    const int half = lane >> 4;                // K-pair selector within fragments

    const float* __restrict__ Arow = valid ? (A + (size_t)(tm * 16 + m) * K) : A;
    const int Ncol4 = Ncol >> 2;

    v8f acc0 = {}, acc1 = {}, acc2 = {}, acc3 = {};

    for (int kc = 0; kc < K; kc += KCHUNK) {
        // ---- cooperative stage: rows [kc,kc+128) x 64 cols, pair-interleaved ----
        {
            const float4* __restrict__ B4 = (const float4*)B;
            #pragma unroll
            for (int it = 0; it < (PAIRROWS * 16) / 256; ++it) {  // 4 iters
                int idx = it * 256 + tid;                         // < 1024
                int r   = idx >> 4;                               // pair-row
                int c4  = idx & 15;                               // column group of 4
                float4 e = B4[(size_t)(kc + 2 * r)     * Ncol4 + tn * 16 + c4]; // even row
                float4 o = B4[(size_t)(kc + 2 * r + 1) * Ncol4 + tn * 16 + c4]; // odd row
                float* p = &ldsB[r * PRSTRIDE + c4 * 8];
                ((float4*)p)[0] = make_float4(e.x, o.x, e.y, o.y);
                ((float4*)p)[1] = make_float4(e.z, o.z, e.w, o.w);
            }
        }
        __syncthreads();

        if (valid) {
            const float* __restrict__ Ap = Arow + kc + half * 2;
            const float* lb0 = &ldsB[half * PRSTRIDE + m * 2];
            #pragma unroll 4
            for (int kk = 0; kk < KCHUNK; kk += 4) {
                // A fragment: lanes 0-15 K=kk,kk+1 ; lanes 16-31 K=kk+2,kk+3
                v2f a;
                a.x = Ap[kk];
                a.y = Ap[kk + 1];
                const float* lb = lb0 + (kk >> 1) * PRSTRIDE;
                v2f b0 = *(const v2f*)(lb +  0);   // cols tile 0 (dword off  m*2)
                v2f b1 = *(const v2f*)(lb + 32);   // cols tile 1 (+16 cols)
                v2f b2 = *(const v2f*)(lb + 64);   // cols tile 2
                v2f b3 = *(const v2f*)(lb + 96);   // cols tile 3
                acc0 = __builtin_amdgcn_wmma_f32_16x16x4_f32(false, a, false, b0, (short)0, acc0, false, false);
                acc1 = __builtin_amdgcn_wmma_f32_16x16x4_f32(false, a, false, b1, (short)0, acc1, false, false);
                acc2 = __builtin_amdgcn_wmma_f32_16x16x4_f32(false, a, false, b2, (short)0, acc2, false, false);
                acc3 = __builtin_amdgcn_wmma_f32_16x16x4_f32(false, a, false, b3, (short)0, acc3, false, false);
            }
        }
        __syncthreads();
    }

    if (valid) {
        // D layout: VGPR r holds row M = r + 8*half, column = lane&15
        #pragma unroll
        for (int r = 0; r < 8; ++r) {
            const size_t row = (size_t)(tm * 16 + r + 8 * half);
            float* __restrict__ Crow = C + row * Ncol + tn * 64 + m;
            Crow[ 0] = acc0[r];
            Crow[16] = acc1[r];
            Crow[32] = acc2[r];
            Crow[48] = acc3[r];
        }
    }
}

// ---------------------------------------------------------------------------
// Edge scatter: one wave per edge; agg[dst] += h[src] * dinv[src]*dinv[dst]
// float4 gathers, 4-wide atomic f32 adds. F % 4 == 0.
// ---------------------------------------------------------------------------
__global__ void scatter_edges(const float* __restrict__ h, float* __restrict__ agg,
                              const int* __restrict__ src, const int* __restrict__ dst,
                              const float* __restrict__ dinv, int E, int F) {
    const int wave = (blockIdx.x * blockDim.x + threadIdx.x) >> 5;
    const int lane = threadIdx.x & 31;
    if (wave >= E) return;
    const int s = src[wave];
    const int d = dst[wave];
    const float coef = dinv[s] * dinv[d];
    const float4* __restrict__ hs = (const float4*)(h + (size_t)s * F);
    float* __restrict__ ad = agg + (size_t)d * F;
    const int q = F >> 2;
    for (int i = lane; i < q; i += 32) {
        float4 v = hs[i];
        atomicAdd(&ad[i * 4 + 0], v.x * coef);
        atomicAdd(&ad[i * 4 + 1], v.y * coef);
        atomicAdd(&ad[i * 4 + 2], v.z * coef);
        atomicAdd(&ad[i * 4 + 3], v.w * coef);
    }
}

// ---------------------------------------------------------------------------
// h = relu( BN( agg + h1*dinv^2 + b1 ) )  written in-place over h1
// ---------------------------------------------------------------------------
__global__ void combine_bn_relu(float* __restrict__ h1, const float* __restrict__ agg,
                                const float* __restrict__ dinv,
                                const float* __restrict__ b1, const float* __restrict__ gamma,
                                const float* __restrict__ beta, const float* __restrict__ mean,
                                const float* __restrict__ var, int n, int F) {
    const int q = F >> 2;
    int idx = blockIdx.x * blockDim.x + threadIdx.x;
    if (idx >= n * q) return;
    const int node = idx / q;
    const int f    = (idx - node * q) * 4;
    const float di = dinv[node];
    const float selfc = di * di;
    float4 hv = ((const float4*)h1)[idx];
    float4 av = ((const float4*)agg)[idx];
    float o[4] = {av.x + hv.x * selfc, av.y + hv.y * selfc,
                  av.z + hv.z * selfc, av.w + hv.w * selfc};
#pragma unroll
    for (int c = 0; c < 4; ++c) {
        float v = o[c] + b1[f + c];
        v = (v - mean[f + c]) * (gamma[f + c] * rsqrtf(var[f + c] + BN_EPS)) + beta[f + c];
        o[c] = fmaxf(v, 0.0f);
    }
    ((float4*)h1)[idx] = make_float4(o[0], o[1], o[2], o[3]);
}

// ---------------------------------------------------------------------------
// out = out(agg2) + h2*dinv^2 + b2
// ---------------------------------------------------------------------------
__global__ void final_add(float* __restrict__ out, const float* __restrict__ h2,
                          const float* __restrict__ dinv, const float* __restrict__ b2,
                          int n, int F) {
    const int q = F >> 2;
    int idx = blockIdx.x * blockDim.x + threadIdx.x;
    if (idx >= n * q) return;
    const int node = idx / q;
    const int f    = (idx - node * q) * 4;
    const float di = dinv[node];
    const float selfc = di * di;
    float4 hv = ((const float4*)h2)[idx];
    float4 ov = ((const float4*)out)[idx];
    ov.x += hv.x * selfc + b2[f + 0];
    ov.y += hv.y * selfc + b2[f + 1];
    ov.z += hv.z * selfc + b2[f + 2];
    ov.w += hv.w * selfc + b2[f + 3];
    ((float4*)out)[idx] = ov;
}

// ---------------------------------------------------------------------------
extern "C" void kernel_launch(void* const* d_in, const int* in_sizes, int n_in,
                              void* d_out, int out_size, void* d_ws, size_t ws_size,
                              hipStream_t stream) {
    (void)n_in; (void)out_size; (void)ws_size;
    const int F_IN = 256, H = 256, C = 64;
    const int N = in_sizes[0] / F_IN;
    const int E = in_sizes[1] / 2;

    const float* x     = (const float*)d_in[0];
    const int*   src   = (const int*)d_in[1];
    const int*   dst   = src + E;
    const float* W1    = (const float*)d_in[2];
    const float* b1    = (const float*)d_in[3];
    const float* gamma = (const float*)d_in[4];
    const float* beta  = (const float*)d_in[5];
    const float* rmean = (const float*)d_in[6];
    const float* rvar  = (const float*)d_in[7];
    const float* W2    = (const float*)d_in[8];
    const float* b2    = (const float*)d_in[9];
    float* out = (float*)d_out;

    char* ws = (char*)d_ws;
    float* h1   = (float*)ws;                                       // N*H floats
    float* agg1 = (float*)(ws + (size_t)N * H * sizeof(float));     // N*H floats (reused as h2)
    float* dinv = (float*)(ws + 2 * (size_t)N * H * sizeof(float)); // N floats
    float* h2   = agg1;

    const int TB = 256;
    // --- degree / normalization ---
    {
        int n4 = N / 4;
        zero_f32v4<<<(n4 + TB - 1) / TB, TB, 0, stream>>>((float4*)dinv, n4);
        deg_count<<<(E + TB - 1) / TB, TB, 0, stream>>>(dst, dinv, E);
        make_dinv<<<(N + TB - 1) / TB, TB, 0, stream>>>(dinv, N);
    }
    // --- layer 1: GEMM (WMMA f32, LDS-staged B) ---
    {
        int rowTiles  = N / 16;
        int blockRows = (rowTiles + 7) / 8;
        int blocks    = blockRows * (H / 64);
        gemm_wmma_f32<<<blocks, TB, 0, stream>>>(x, W1, h1, N, F_IN, H);
    }
    // --- layer 1: aggregation ---
    {
        int n4 = (int)(((size_t)N * H) / 4);
        zero_f32v4<<<(n4 + TB - 1) / TB, TB, 0, stream>>>((float4*)agg1, n4);
        int blocks = (E * 32 + TB - 1) / TB;   // one wave per edge
        scatter_edges<<<blocks, TB, 0, stream>>>(h1, agg1, src, dst, dinv, E, H);
    }
    // --- combine + BN + ReLU (in-place into h1) ---
    {
        int total = (int)(((size_t)N * H) / 4);
        combine_bn_relu<<<(total + TB - 1) / TB, TB, 0, stream>>>(
            h1, agg1, dinv, b1, gamma, beta, rmean, rvar, N, H);
    }
    // --- layer 2: GEMM (WMMA f32) into h2 (= agg1 buffer) ---
    {
        int rowTiles  = N / 16;
        int blockRows = (rowTiles + 7) / 8;
        int blocks    = blockRows * (C / 64);
        gemm_wmma_f32<<<blocks, TB, 0, stream>>>(h1, W2, h2, N, H, C);
    }
    // --- layer 2: aggregation into d_out ---
    {
        int n4 = (int)(((size_t)N * C) / 4);
        zero_f32v4<<<(n4 + TB - 1) / TB, TB, 0, stream>>>((float4*)out, n4);
        int blocks = (E * 32 + TB - 1) / TB;
        scatter_edges<<<blocks, TB, 0, stream>>>(h2, out, src, dst, dinv, E, C);
    }
    // --- final: out += h2*dinv^2 + b2 ---
    {
        int total = (int)(((size_t)N * C) / 4);
        final_add<<<(total + TB - 1) / TB, TB, 0, stream>>>(out, h2, dinv, b2, N, C);
    }
}